// TypedBinaryTreeLSTMLayer_79783312490657
// MI455X (gfx1250) — compile-verified
//
#include <hip/hip_runtime.h>

typedef float v2f __attribute__((ext_vector_type(2)));
typedef float v8f __attribute__((ext_vector_type(8)));

// Problem constants (match the reference)
constexpr int Bb  = 32;    // batch
constexpr int Nn  = 9;     // N
constexpr int Mm  = 128;   // M
constexpr int Vv  = 512;   // V
constexpr int Hh  = 256;   // H
constexpr int Kk  = 8;     // K
constexpr int NTt = 12;    // TYPES-9
constexpr int NP1 = 10;    // N+1
constexpr int LR  = 80;    // K*(N+1) logits rows

__global__ __launch_bounds__(1024)
void ttree_fused(const float* __restrict__ dec,    // (B,N,M,V)
                 const float* __restrict__ temb,   // (21,H)
                 const float* __restrict__ W,      // (8,12,80,H)
                 const float* __restrict__ bias,   // (8,12,80)
                 const float* __restrict__ gum,    // (B,K,10)
                 const int*   __restrict__ tty,    // (B,)
                 const int*   __restrict__ spn,    // (B,)
                 float* __restrict__ out_d,        // (B,M,V)
                 float* __restrict__ out_v)        // (B,20,3,V)
{
    __shared__ __align__(16) float s_emb[Hh];
    __shared__ float s_logits[LR];
    __shared__ int   s_sel[Kk];
    __shared__ float s_hot[Kk];
    __shared__ int   s_rawlen[Kk];
    __shared__ int   s_start[Kk];
    __shared__ int   s_end[Kk];
    __shared__ int   s_rowk[Mm];

    const int b    = blockIdx.x;
    const int tid  = threadIdx.x;
    const int lane = tid & 31;
    const int wave = tid >> 5;
    const int tt   = tty[b];
    const int sp   = spn[b];           // span in [2,9]; valid slots j = 0..sp

    if (tid < Hh) s_emb[tid]   = temb[tt * Hh + tid];
    if (tid < Kk) s_rawlen[tid] = 0;
    __syncthreads();

    // ---------------- Phase 1: logits = Wi * emb + bi via V_WMMA_F32_16X16X4_F32 (wave 0) ----------------
    if (wave == 0) {
        const size_t widx = (size_t)(sp - 2) * NTt + (tt - 9);   // t==20 -> index 11, still in range
        const float* Wp = W    + widx * LR * Hh;
        const float* bp = bias + widx * LR;
        const int half = lane >> 4;        // 0: K pair {0,1}, 1: K pair {2,3}
        const int lrow = lane & 15;        // A-matrix row within tile
        const v2f* e2 = (const v2f*)s_emb;
        for (int t = 0; t < 5; ++t) {      // 80 rows = 5 x 16-row tiles (exact)
            const v2f* w2 = (const v2f*)(Wp + (size_t)(t * 16 + lrow) * Hh);
            v8f c = {0.f,0.f,0.f,0.f,0.f,0.f,0.f,0.f};
            #pragma unroll 8
            for (int k2 = 0; k2 < Hh / 2; k2 += 2) {   // k2 = k0/2; K step of 4 floats
                v2f a  = w2[k2 + half];                // W[row, k0+2*half .. +1]
                v2f bb = e2[k2 + half];                // emb replicated across N columns
                c = __builtin_amdgcn_wmma_f32_16x16x4_f32(
                        false, a, false, bb, (short)0, c, false, false);
            }
            if (lrow == 0) {                           // lanes 0 and 16 hold column N=0
                #pragma unroll
                for (int r = 0; r < 8; ++r) {          // D: VGPR r -> row r + 8*half
                    const int o = t * 16 + 8 * half + r;
                    s_logits[o] = c[r] + bp[o];
                }
            }
        }
    }
    __syncthreads();

    // ---------------- Phase 2: masked log_softmax + gumbel + hard selection (threads 0..7) ----------------
    if (tid < Kk) {
        const int k = tid;
        int   sel;
        float hot;
        if (tt == 20) {
            // fixed start template: row0 -> slot 1 (decoding 0), rows 1.. -> slot 0 (pad)
            sel = (k == 0) ? 1 : 0;
            hot = 1.0f;
        } else {
            const int nv = sp + 1;                      // valid slots 0..sp
            float lg[NP1];
            float mx = -__builtin_inff();
            for (int j = 0; j < nv; ++j) {
                lg[j] = s_logits[k * NP1 + j];
                mx = fmaxf(mx, lg[j]);
            }
            float se = 0.f;
            for (int j = 0; j < nv; ++j) se += expf(lg[j] - mx);
            const float lse = logf(se);
            float z[NP1];
            float mz = -__builtin_inff();
            for (int j = 0; j < nv; ++j) {
                z[j] = (lg[j] - mx - lse) + gum[((size_t)b * Kk + k) * NP1 + j];  // TAU=1
                mz = fmaxf(mz, z[j]);
            }
            float s2 = 0.f;
            int   am = 0;
            float bv = -__builtin_inff();
            for (int j = 0; j < nv; ++j) {
                const float e = expf(z[j] - mz);
                s2 += e;
                if (e > bv) { bv = e; am = j; }         // lowest index wins ties
            }
            const float ysoft = bv / s2;                // y_soft at the argmax
            sel = am;
            hot = (1.0f + ysoft) - ysoft;               // straight-through forward value
        }
        s_sel[k] = sel;
        s_hot[k] = hot;
    }
    __syncthreads();

    // ---------------- Phase 3: raw_len[k] via per-(k,m) argmax over V (all 32 waves) ----------------
    {
        const int k   = wave >> 2;                      // 4 waves per k-slot
        const int sel = s_sel[k];
        if (sel > 0) {                                  // sel==0 -> pad row -> raw_len stays 0
            const int n = sel - 1;
            const float* base = dec + ((size_t)b * Nn + n) * Mm * Vv;
            int local = 0;
            for (int mi = 0; mi < 32; ++mi) {
                const int m = ((wave & 3) << 5) + mi;
                if (m + 1 < Mm)
                    __builtin_prefetch(base + (size_t)(m + 1) * Vv + lane * 16, 0, 0);
                const float4* r4 = (const float4*)(base + (size_t)m * Vv);
                float bv = -__builtin_inff();
                int   bi = 0;
                #pragma unroll
                for (int j = 0; j < 4; ++j) {
                    const int q  = lane + (j << 5);     // coalesced 512B per step
                    const float4 x = r4[q];
                    const int i0 = q << 2;
                    if (x.x > bv) { bv = x.x; bi = i0;     }
                    if (x.y > bv) { bv = x.y; bi = i0 + 1; }
                    if (x.z > bv) { bv = x.z; bi = i0 + 2; }
                    if (x.w > bv) { bv = x.w; bi = i0 + 3; }
                }
                #pragma unroll
                for (int off = 16; off >= 1; off >>= 1) {   // wave32 argmax reduce
                    const float ov = __shfl_xor(bv, off, 32);
                    const int   oi = __shfl_xor(bi, off, 32);
                    if (ov > bv || (ov == bv && oi < bi)) { bv = ov; bi = oi; }
                }
                if (bi != 0) local = max(local, m + 1); // idx != PAD extends raw length
            }
            if (lane == 0) atomicMax(&s_rawlen[k], local);
        }
    }
    __syncthreads();

    // ---------------- Phase 4: sequential clipped concat intervals ----------------
    if (tid == 0) {
        int e = 0;
        for (int k = 0; k < Kk; ++k) {
            s_start[k] = e;
            e = min(e + s_rawlen[k], Mm);
            s_end[k] = e;
        }
    }
    __syncthreads();
    if (tid < Mm) {
        int kk = -1;
        for (int k = 0; k < Kk; ++k)
            if (tid >= s_start[k] && tid < s_end[k]) { kk = k; break; }
        s_rowk[tid] = kk;
    }
    __syncthreads();

    // ---------------- Phase 5: gather + scale write of new_d, zero new_v ----------------
    float4* od4 = (float4*)(out_d + (size_t)b * Mm * Vv);
    for (int idx = tid; idx < Mm * (Vv / 4); idx += 1024) {
        const int p  = idx >> 7;                        // output row (V/4 = 128 float4/row)
        const int c4 = idx & 127;
        const int kk = s_rowk[p];
        float4 val = {0.f, 0.f, 0.f, 0.f};
        if (kk >= 0) {
            const int n = s_sel[kk] - 1;                // lens>0 implies sel>0
            const int m = p - s_start[kk];
            const float4* src =
                (const float4*)(dec + (((size_t)b * Nn + n) * Mm + m) * Vv);
            const float4 x = src[c4];
            const float  h = s_hot[kk];
            val.x = x.x * h; val.y = x.y * h; val.z = x.z * h; val.w = x.w * h;
        }
        od4[idx] = val;
    }
    float4* ov4 = (float4*)(out_v + (size_t)b * 20 * 3 * Vv);
    const float4 z4 = {0.f, 0.f, 0.f, 0.f};
    for (int idx = tid; idx < 20 * 3 * (Vv / 4); idx += 1024) ov4[idx] = z4;
}

extern "C" void kernel_launch(void* const* d_in, const int* in_sizes, int n_in,
                              void* d_out, int out_size, void* d_ws, size_t ws_size,
                              hipStream_t stream) {
    (void)in_sizes; (void)n_in; (void)d_ws; (void)ws_size; (void)out_size;
    const float* dec  = (const float*)d_in[0];   // decodings
    const float* temb = (const float*)d_in[1];   // type_emb
    const float* W    = (const float*)d_in[2];   // dec_sem_W
    const float* bia  = (const float*)d_in[3];   // dec_sem_b
    const float* gum  = (const float*)d_in[4];   // gumbel
    const int*   tty  = (const int*)d_in[5];     // target_types
    const int*   spn  = (const int*)d_in[6];     // spans

    float* out_d = (float*)d_out;                         // (B,M,V)
    float* out_v = out_d + (size_t)Bb * Mm * Vv;          // (B,20,3,V)

    ttree_fused<<<dim3(Bb), dim3(1024), 0, stream>>>(
        dec, temb, W, bia, gum, tty, spn, out_d, out_v);
}